// GCN_30562987278370
// MI455X (gfx1250) — compile-verified
//
#include <hip/hip_runtime.h>

// GCN 3-layer forward for gfx1250 (MI455X).
// GEMMs use V_WMMA_F32_16X16X4_F32 (fp32 matrix core path, wave32).
// Scatter phase: one wave per edge, float4 per lane, global_atomic_add_f32
// (accumulator + gathered features fit in 192MB L2, so atomics resolve there).

typedef __attribute__((ext_vector_type(2))) float v2f;
typedef __attribute__((ext_vector_type(8))) float v8f;

#define D 128  // feature dim for all layers

// ---------- degree / normalization ----------
__global__ void k_deg_init(float* __restrict__ deg, int n) {
  int i = blockIdx.x * blockDim.x + threadIdx.x;
  if (i < n) deg[i] = 1.0f;  // self-loop contributes 1 to every node's degree
}

__global__ void k_deg_count(const int* __restrict__ dst, float* __restrict__ deg, int e) {
  int i = blockIdx.x * blockDim.x + threadIdx.x;
  if (i < e) atomicAdd(&deg[dst[i]], 1.0f);
}

__global__ void k_deg_rsqrt(float* __restrict__ deg, int n) {
  int i = blockIdx.x * blockDim.x + threadIdx.x;
  if (i < n) deg[i] = rsqrtf(deg[i]);  // deg >= 1 always (self loops)
}

// ---------- bias broadcast (accumulator init) ----------
__global__ void k_bias_init(float* __restrict__ out, const float* __restrict__ b, int total) {
  int i = blockIdx.x * blockDim.x + threadIdx.x;
  if (i < total) out[i] = b[i & (D - 1)];
}

// ---------- dense GEMM: H[nrows x 128] = act(A[nrows x 128]) @ W[128 x 128] ----------
// Block: 256 threads = 8 wave32s. Each wave computes a 16x128 output stripe
// as 8 WMMA 16x16 tiles, K-loop in steps of 4 (V_WMMA_F32_16X16X4_F32).
template <bool RELU_IN>
__global__ __launch_bounds__(256) void k_gemm128(const float* __restrict__ A,
                                                 const float* __restrict__ W,
                                                 float* __restrict__ H, int nrows) {
  __shared__ float Ws[D * D];  // 64 KB, loaded once per block

  const int tid = threadIdx.x;
  {
    const float4* __restrict__ Wv = (const float4*)W;
    float4* Sv = (float4*)Ws;
#pragma unroll
    for (int i = 0; i < (D * D / 4) / 256; ++i)  // 16 iters
      Sv[tid + i * 256] = Wv[tid + i * 256];
  }
  __syncthreads();

  const int wave = tid >> 5;
  const int lane = tid & 31;
  const int row0 = blockIdx.x * 128 + wave * 16;
  if (row0 >= nrows) return;  // wave-uniform (nrows % 16 == 0): EXEC stays all-1s

  const int m  = lane & 15;          // row (A) / col (B,C,D) within tile
  const int kb = (lane >> 4) << 1;   // K sub-offset: 0 for lanes 0-15, 2 for 16-31

  v8f acc[8] = {};

  const float* __restrict__ Arow = A + (size_t)(row0 + m) * D;
  for (int k = 0; k < D; k += 4) {
    float ax = Arow[k + kb];
    float ay = Arow[k + kb + 1];
    if (RELU_IN) {
      ax = fmaxf(ax, 0.0f);
      ay = fmaxf(ay, 0.0f);
    }
    v2f a;
    a.x = ax;
    a.y = ay;
#pragma unroll
    for (int t = 0; t < 8; ++t) {
      const int col = t * 16 + m;
      v2f b;
      b.x = Ws[(k + kb) * D + col];
      b.y = Ws[(k + kb + 1) * D + col];
      acc[t] = __builtin_amdgcn_wmma_f32_16x16x4_f32(
          /*neg_a=*/false, a, /*neg_b=*/false, b,
          /*c_mod=*/(short)0, acc[t], /*reuse_a=*/false, /*reuse_b=*/false);
    }
  }

  // C/D layout: VGPR v -> M = v + 8*(lane>=16), N = lane&15
  const int rb = row0 + ((lane >> 4) << 3);
#pragma unroll
  for (int t = 0; t < 8; ++t) {
    const int col = t * 16 + m;
#pragma unroll
    for (int v = 0; v < 8; ++v)
      H[(size_t)(rb + v) * D + col] = acc[t][v];
  }
}

// ---------- edge scatter: acc[dst] += h[src] * dis[src]*dis[dst] ----------
// One wave per edge (8 waves / 256-thread block); lane handles 4 floats.
// Edges [0,E) come from edge_index; edges [E, E+N) are self loops.
__global__ __launch_bounds__(256) void k_scatter(const float* __restrict__ h,
                                                 const int* __restrict__ ei, int E, int n,
                                                 const float* __restrict__ dis,
                                                 float* __restrict__ acc) {
  const int wid  = blockIdx.x * (blockDim.x >> 5) + (threadIdx.x >> 5);
  const int lane = threadIdx.x & 31;
  const int total = E + n;
  if (wid >= total) return;

  int s, d;
  if (wid < E) {
    s = ei[wid];
    d = ei[E + wid];
  } else {
    s = d = wid - E;
  }
  const float nrm = dis[s] * dis[d];

  const float4 hv = *(const float4*)(h + (size_t)s * D + lane * 4);
  float* o = acc + (size_t)d * D + lane * 4;
  atomicAdd(o + 0, hv.x * nrm);
  atomicAdd(o + 1, hv.y * nrm);
  atomicAdd(o + 2, hv.z * nrm);
  atomicAdd(o + 3, hv.w * nrm);
}

extern "C" void kernel_launch(void* const* d_in, const int* in_sizes, int n_in,
                              void* d_out, int out_size, void* d_ws, size_t ws_size,
                              hipStream_t stream) {
  const float* x  = (const float*)d_in[0];
  const int*   ei = (const int*)d_in[1];   // [2, E] row-major: src row then dst row
  const float* W1 = (const float*)d_in[2];
  const float* b1 = (const float*)d_in[3];
  const float* W2 = (const float*)d_in[4];
  const float* b2 = (const float*)d_in[5];
  const float* W3 = (const float*)d_in[6];
  const float* b3 = (const float*)d_in[7];

  const int N = in_sizes[0] / D;
  const int E = in_sizes[1] / 2;

  // Workspace layout: deg/dis [N, padded], buf1 [N*128], buf2 [N*128]  (~103 MB)
  float* deg  = (float*)d_ws;
  size_t npad = ((size_t)N + 255) & ~(size_t)255;
  float* buf1 = deg + npad;
  float* buf2 = buf1 + (size_t)N * D;
  float* out  = (float*)d_out;

  const dim3 blk(256);
  const int gN    = (N + 255) / 256;
  const int gE    = (E + 255) / 256;
  const int totF  = N * D;
  const int gF    = (totF + 255) / 256;
  const int gGemm = (N + 127) / 128;
  const int gScat = ((E + N) + 7) / 8;  // 8 waves per block, 1 wave per edge

  // Symmetric normalization: dis = rsqrt(1 + indegree)
  k_deg_init<<<gN, blk, 0, stream>>>(deg, N);
  k_deg_count<<<gE, blk, 0, stream>>>(ei + E, deg, E);
  k_deg_rsqrt<<<gN, blk, 0, stream>>>(deg, N);

  // Layer 1: h = x @ W1 ; out1 = scatter(h) + b1
  k_gemm128<false><<<gGemm, blk, 0, stream>>>(x, W1, buf1, N);
  k_bias_init<<<gF, blk, 0, stream>>>(buf2, b1, totF);
  k_scatter<<<gScat, blk, 0, stream>>>(buf1, ei, E, N, deg, buf2);

  // Layer 2: h = relu(out1) @ W2 ; out2 = scatter(h) + b2
  k_gemm128<true><<<gGemm, blk, 0, stream>>>(buf2, W2, buf1, N);
  k_bias_init<<<gF, blk, 0, stream>>>(buf2, b2, totF);
  k_scatter<<<gScat, blk, 0, stream>>>(buf1, ei, E, N, deg, buf2);

  // Layer 3: h = relu(out2) @ W3 ; out = scatter(h) + b3
  k_gemm128<true><<<gGemm, blk, 0, stream>>>(buf2, W3, buf1, N);
  k_bias_init<<<gF, blk, 0, stream>>>(out, b3, totF);
  k_scatter<<<gScat, blk, 0, stream>>>(buf1, ei, E, N, deg, out);
}